// ScRRAMBLeMLPMixerBlock_19164144074964
// MI455X (gfx1250) — compile-verified
//
#include <hip/hip_runtime.h>
#include <hip/hip_bf16.h>
#include <math.h>

// ScRRAMBLe MLP-mixer block for MI455X (gfx1250, wave32).
// Bandwidth-bound problem: stream Wi (512MB) + Co (256MB) once; the routing
// einsum is a (8192x8192)@(8192x64) fp32 GEMM done with V_WMMA_F32_16X16X4_F32.

#define CI   2048
#define CO   2048
#define SLOT 64
#define SPC  4
#define NEL  (CI * SPC * SLOT)        // 524288 (flattened y / out size)
#define KTOT (CI * SPC)               // 8192 reduction dim for routing GEMM

typedef __attribute__((ext_vector_type(2))) float v2f;
typedef __attribute__((ext_vector_type(4))) float v4f;
typedef __attribute__((ext_vector_type(8))) float v8f;

__device__ __forceinline__ float gelu_tanh(float x) {
    // jax.nn.gelu approximate=True: 0.5*x*(1+tanh(sqrt(2/pi)*(x+0.044715*x^3)))
    float x3 = x * x * x;
    float t  = tanhf(0.7978845608028654f * (x + 0.044715f * x3));
    return 0.5f * x * (1.0f + t);
}

// ---------------------------------------------------------------------------
// Phase 1: per-capsule block matvec  y[i,l,j] = sum_{k,m} Wi[i,j,k,l,m]*x[i,m,k]
// then gelu; write raw y (flat [i*256 + l*64 + j]) and per-block sum/sumsq.
// Wi flat: i*65536 + j*1024 + k*16 + l*4 + m  -> thread (j,l) reads a float4
// per k (16B aligned); groups of 4 threads cover 64 contiguous bytes.
// ---------------------------------------------------------------------------
__global__ __launch_bounds__(256) void phase1_capsule_matvec(
    const float* __restrict__ x, const float* __restrict__ Wi,
    float* __restrict__ yraw, float* __restrict__ partials) {
    __shared__ float lx[256];
    __shared__ float red1[256];
    __shared__ float red2[256];

    const int i = blockIdx.x;
    const int t = threadIdx.x;

    lx[t] = x[(size_t)i * 256 + t];   // xr[i, m, k] at m*64 + k
    __syncthreads();

    const int j = t >> 2;             // output slot (0..63)
    const int l = t & 3;              // output spc  (0..3)

    const v4f* wrow = (const v4f*)(Wi + (size_t)i * 65536 + (size_t)j * 1024 + (size_t)l * 4);
    float acc = 0.0f;
#pragma unroll 8
    for (int k = 0; k < 64; ++k) {
        v4f w4 = __builtin_nontemporal_load(&wrow[k * 4]);  // Wi[i,j,k,l,0..3]
        acc = fmaf(w4.x, lx[k],        acc);
        acc = fmaf(w4.y, lx[64  + k],  acc);
        acc = fmaf(w4.z, lx[128 + k],  acc);
        acc = fmaf(w4.w, lx[192 + k],  acc);
    }

    float g = gelu_tanh(acc);
    yraw[(size_t)i * 256 + (size_t)l * 64 + j] = g;

    red1[t] = g;
    red2[t] = g * g;
    __syncthreads();
#pragma unroll
    for (int s = 128; s > 0; s >>= 1) {
        if (t < s) { red1[t] += red1[t + s]; red2[t] += red2[t + s]; }
        __syncthreads();
    }
    if (t == 0) {
        partials[2 * i]     = red1[0];
        partials[2 * i + 1] = red2[0];
    }
}

// ---------------------------------------------------------------------------
// Phase 2a: reduce 2048 (sum, sumsq) partials -> mu, rsigma (deterministic tree)
// ---------------------------------------------------------------------------
__global__ __launch_bounds__(256) void phase2a_ln_stats(
    const float* __restrict__ partials, float* __restrict__ stats) {
    __shared__ float s1[256];
    __shared__ float s2[256];
    const int t = threadIdx.x;
    float a = 0.0f, b = 0.0f;
#pragma unroll
    for (int p = t; p < 2048; p += 256) { a += partials[2 * p]; b += partials[2 * p + 1]; }
    s1[t] = a; s2[t] = b;
    __syncthreads();
#pragma unroll
    for (int s = 128; s > 0; s >>= 1) {
        if (t < s) { s1[t] += s1[t + s]; s2[t] += s2[t + s]; }
        __syncthreads();
    }
    if (t == 0) {
        const float N  = (float)NEL;
        float mu  = s1[0] / N;
        float var = s2[0] / N - mu * mu;
        stats[0] = mu;
        stats[1] = rsqrtf(var + 1e-6f);
    }
}

// ---------------------------------------------------------------------------
// Phase 2b: apply global LayerNorm -> ynorm[c*64+m]  (c = i*4+spc, m = slot)
// ---------------------------------------------------------------------------
__global__ __launch_bounds__(256) void phase2b_ln_apply(
    const float* __restrict__ yraw, const float* __restrict__ ln_scale,
    const float* __restrict__ ln_bias, const float* __restrict__ stats,
    float* __restrict__ ynorm) {
    const int idx = blockIdx.x * 256 + threadIdx.x;
    const float mu = stats[0];
    const float rs = stats[1];
    ynorm[idx] = (yraw[idx] - mu) * rs * ln_scale[idx] + ln_bias[idx];
}

// ---------------------------------------------------------------------------
// Phase 3: routing GEMM with fp32 WMMA.
//   out[r, m] = sum_c CoT[c, r] * ynorm[c, m],  r=(k*4+l) 0..8191, m 0..63
// CoT is row-major in memory (Co flat = c*8192 + r). Compute D = A x B + C with
// A = ynorm^T tile (16m x 4c), B = CoT tile (4c x 16r), per ISA lane layouts:
//   A: lane(l): row M = l%16, VGPR v: K = (l/16)*2 + v
//   B: lane(l): col N = l%16, VGPR v: K = (l/16)*2 + v
//   D: lane(l): col N = l%16, VGPR v: row M = v + 8*(l/16)
// Block = 4 waves; wave w owns m-tile w (all 4 tiles -> Co streamed once/block).
// ---------------------------------------------------------------------------
__global__ __launch_bounds__(128) void phase3_routing_wmma(
    const float* __restrict__ Co, const float* __restrict__ ynorm,
    float* __restrict__ out) {
    const int r0   = blockIdx.x * 16;
    const int wave = threadIdx.x >> 5;
    const int lane = threadIdx.x & 31;
    const int half = lane >> 4;   // 0 or 1
    const int lid  = lane & 15;
    const int m0   = wave * 16;

    const int mA = m0 + lid;      // A-matrix row (m index)
    const int rB = r0 + lid;      // B-matrix col (r index)

    v8f acc = {};
    for (int c0 = 0; c0 < KTOT; c0 += 4) {
        const int cA = c0 + half * 2;
        v2f a, b;
        a.x = ynorm[(size_t)cA * 64 + mA];
        a.y = ynorm[(size_t)(cA + 1) * 64 + mA];
        b.x = __builtin_nontemporal_load(&Co[(size_t)cA * 8192 + rB]);
        b.y = __builtin_nontemporal_load(&Co[(size_t)(cA + 1) * 8192 + rB]);
        acc = __builtin_amdgcn_wmma_f32_16x16x4_f32(
            /*neg_a=*/false, a, /*neg_b=*/false, b,
            /*c_mod=*/(short)0, acc, /*reuse_a=*/false, /*reuse_b=*/false);
    }

    // D[M,N] = out[r0+N, m0+M]; out flat = r*64 + m
#pragma unroll
    for (int v = 0; v < 8; ++v) {
        out[(size_t)(r0 + lid) * 64 + (m0 + v + 8 * half)] = acc[v];
    }
}

extern "C" void kernel_launch(void* const* d_in, const int* in_sizes, int n_in,
                              void* d_out, int out_size, void* d_ws, size_t ws_size,
                              hipStream_t stream) {
    const float* x        = (const float*)d_in[0];
    const float* Wi       = (const float*)d_in[1];
    const float* ln_scale = (const float*)d_in[2];
    const float* ln_bias  = (const float*)d_in[3];
    const float* Co       = (const float*)d_in[4];
    float* out = (float*)d_out;

    // Workspace layout (floats): [ynorm: NEL][partials: 4096][stats: 2]
    float* ws       = (float*)d_ws;
    float* ynorm    = ws;
    float* partials = ws + NEL;
    float* stats    = ws + NEL + 4096;
    float* yraw     = out;   // reuse d_out as temp for raw gelu(y); phase3 overwrites

    phase1_capsule_matvec<<<dim3(CI), dim3(256), 0, stream>>>(x, Wi, yraw, partials);
    phase2a_ln_stats<<<dim3(1), dim3(256), 0, stream>>>(partials, stats);
    phase2b_ln_apply<<<dim3(NEL / 256), dim3(256), 0, stream>>>(yraw, ln_scale, ln_bias, stats, ynorm);
    phase3_routing_wmma<<<dim3(CO * SPC / 16), dim3(128), 0, stream>>>(Co, ynorm, out);
}